// SingleSimplePatch_91233695302148
// MI455X (gfx1250) — compile-verified
//
#include <hip/hip_runtime.h>

typedef float v2f __attribute__((ext_vector_type(2)));
typedef float v8f __attribute__((ext_vector_type(8)));

#define GH 47
#define GW 47
#define LWIN 2209            // 47*47 windows per image
#define BATCH 32
#define KSEL 16
#define NPATCH (BATCH * KSEL)
#define INS_N (3 * 66 * 66)    // padded input patch in LDS   (13068 f32)
#define Y1_N  (32 * 34 * 34)   // padded conv1 out, pair-interleaved (36992 f32)
#define W2P_N (288 * 64)       // reordered conv2 weights      (18432 f32)

// ---------------- kernel 1: 16x16/stride-8 window variances ----------------
__global__ void var_kernel(const float* __restrict__ x, float* __restrict__ var) {
    int wid  = blockIdx.x * 8 + (threadIdx.x >> 5);
    int lane = threadIdx.x & 31;
    if (wid >= BATCH * LWIN) return;            // wave-uniform exit
    int b  = wid / LWIN;
    int r  = wid - b * LWIN;
    int gy = r / GW;
    int gx = r - gy * GW;
    int h0 = gy * 8, w0 = gx * 8;
    const float* xb = x + (size_t)b * 3 * 384 * 384;
    float s = 0.f, s2 = 0.f;
    for (int e = lane; e < 768; e += 32) {
        int ic  = e >> 8;
        int rem = e & 255;
        int rr  = rem >> 4;
        int cc  = rem & 15;
        float v = xb[(size_t)ic * 147456 + (size_t)(h0 + rr) * 384 + (w0 + cc)];
        s  += v;
        s2 += v * v;
    }
    for (int off = 16; off >= 1; off >>= 1) {
        s  += __shfl_down(s,  off, 32);
        s2 += __shfl_down(s2, off, 32);
    }
    if (lane == 0) {
        var[wid] = (s2 - s * s * (1.0f / 768.0f)) * (1.0f / 767.0f);
    }
}

// ---------------- kernel 2: top-16 per batch + faithful (buggy) decode ------
__global__ void topk_kernel(const float* __restrict__ var, int* __restrict__ coords) {
    __shared__ float vals[LWIN];
    __shared__ float rv[256];
    __shared__ int   ri[256];
    int b = blockIdx.x;
    int t = threadIdx.x;
    for (int i = t; i < LWIN; i += 256) vals[i] = var[b * LWIN + i];
    __syncthreads();
    for (int it = 0; it < KSEL; ++it) {
        float bv = -3.402823e38f;
        int   bi = LWIN;
        for (int i = t; i < LWIN; i += 256) {
            float v = vals[i];
            if (v > bv) { bv = v; bi = i; }
        }
        rv[t] = bv; ri[t] = bi;
        __syncthreads();
        for (int sft = 128; sft > 0; sft >>= 1) {
            if (t < sft) {
                float ov = rv[t + sft];
                int   oi = ri[t + sft];
                if (ov > rv[t] || (ov == rv[t] && oi < ri[t])) { rv[t] = ov; ri[t] = oi; }
            }
            __syncthreads();
        }
        if (t == 0) {
            int idx = ri[0];
            int hh  = (idx / 48) * 8;           // faithful: divisor W//8 = 48
            int ww  = (idx % 48) * 8;
            if (hh > 320) hh = 320;             // dynamic_slice clamp (384-64)
            if (ww > 320) ww = 320;
            coords[(b * KSEL + it) * 2 + 0] = hh;
            coords[(b * KSEL + it) * 2 + 1] = ww;
            vals[idx] = -3.402823e38f;
        }
        __syncthreads();
    }
}

// ---------------- kernel 3: per-patch encoder (conv1 VALU, conv2 WMMA f32) --
// conv2 reduction order: k' = (ky*3+kx)*32 + ic  (channel-fastest per tap)
// y1 stored channel-pair interleaved: y1p[((ic>>1)*1156 + row*34 + col)*2 + (ic&1)]
// w2 staged in LDS pair-interleaved: w2p[(pp*64 + oc)*2 + (k'&1)], pp = k'>>1
__global__ __launch_bounds__(256) void encode_kernel(
    const float* __restrict__ x,  const float* __restrict__ w1,
    const float* __restrict__ b1, const float* __restrict__ w2,
    const float* __restrict__ b2, const int* __restrict__ coords,
    float* __restrict__ feats) {
    extern __shared__ float smem[];
    float* in_s = smem;              // [3][66][66] zero-padded patch
    float* y1p  = in_s + INS_N;      // pair-interleaved padded conv1 out
    float* w2p  = y1p + Y1_N;        // reordered conv2 weights
    float* w1_s = w2p + W2P_N;       // 864
    float* b1_s = w1_s + 864;        // 32
    float* gap  = b1_s + 32;         // [8 waves][64 oc]

    int patch = blockIdx.x;
    int b     = patch >> 4;
    int t     = threadIdx.x;
    int h0    = coords[patch * 2 + 0];
    int w0    = coords[patch * 2 + 1];
    const float* xb = x + (size_t)b * 3 * 384 * 384;

    for (int i = t; i < INS_N; i += 256) in_s[i] = 0.f;
    for (int i = t; i < Y1_N;  i += 256) y1p[i]  = 0.f;
    for (int i = t; i < 864;   i += 256) w1_s[i] = w1[i];
    if (t < 32) b1_s[t] = b1[t];
    __syncthreads();

    // fill interior of zero-padded patch
    for (int i = t; i < 3 * 64 * 64; i += 256) {
        int ic  = i >> 12;
        int rem = i & 4095;
        int iy  = rem >> 6, ix = rem & 63;
        in_s[ic * 4356 + (iy + 1) * 66 + (ix + 1)] =
            xb[(size_t)ic * 147456 + (size_t)(h0 + iy) * 384 + (w0 + ix)];
    }
    // stage + reorder w2 -> w2p  (k' = (ky*3+kx)*32 + ic, pair-interleaved)
    for (int i = t; i < W2P_N; i += 256) {
        int oc  = i / 288;
        int rem = i - oc * 288;
        int ic  = rem / 9;
        int r   = rem - ic * 9;              // ky*3+kx
        int kp  = r * 32 + ic;
        w2p[((kp >> 1) * 64 + oc) * 2 + (kp & 1)] = w2[i];
    }
    __syncthreads();

    // conv1 3->32, 3x3, s2, p1 + ReLU -> y1p interior
    for (int o = t; o < 32768; o += 256) {
        int oc = o >> 10;
        int p  = o & 1023;
        int oy = p >> 5, ox = p & 31;
        float acc = b1_s[oc];
        const float* wrow = &w1_s[oc * 27];
        int base = (2 * oy) * 66 + 2 * ox;
        for (int ic = 0; ic < 3; ++ic) {
            const float* ip = &in_s[ic * 4356 + base];
            const float* wp = wrow + ic * 9;
            for (int ky = 0; ky < 3; ++ky)
                for (int kx = 0; kx < 3; ++kx)
                    acc += ip[ky * 66 + kx] * wp[ky * 3 + kx];
        }
        y1p[((oc >> 1) * 1156 + (oy + 1) * 34 + (ox + 1)) * 2 + (oc & 1)] =
            acc > 0.f ? acc : 0.f;
    }
    __syncthreads();

    // conv2 32->64, 3x3, s2, p1 as GEMM: [256 pos x 288] x [288 x 64]
    // fp32 WMMA 16x16x4; per wave: 2 M-tiles x 4 N-tiles, 72 k-steps each
    int wv    = t >> 5;
    int lane  = t & 31;
    int n_l   = lane & 15;   // B column / A row-in-tile (ISA: both = lane&15)
    int mhalf = lane >> 4;   // lanes 16-31 carry K = 2,3 halves
    for (int nt = 0; nt < 4; ++nt) {
        float psum = 0.f;
        int ocidx = nt * 16 + n_l;
        for (int mi = 0; mi < 2; ++mi) {
            int mt = wv * 2 + mi;
            int M  = mt * 16 + n_l;          // output position index 0..255
            int py = M >> 4, px = M & 15;
            int rb = (2 * py) * 34 + 2 * px; // padded y1 window top-left
            v8f c = {0.f, 0.f, 0.f, 0.f, 0.f, 0.f, 0.f, 0.f};
#pragma unroll
            for (int g = 0; g < 9; ++g) {    // (ky,kx) tap
                int aBase = rb + (g / 3) * 34 + (g % 3);   // compile-time adds
#pragma unroll
                for (int j = 0; j < 8; ++j) {
                    int cp = 2 * j + mhalf;  // channel-pair index (ic = 2*cp)
                    v2f a  = *(const v2f*)&y1p[(cp * 1156 + aBase) * 2];
                    int pp = g * 16 + 2 * j + mhalf;
                    v2f bb = *(const v2f*)&w2p[(pp * 64 + ocidx) * 2];
                    c = __builtin_amdgcn_wmma_f32_16x16x4_f32(
                            false, a, false, bb, (short)0, c, false, false);
                }
            }
            for (int r = 0; r < 8; ++r) psum += c[r];  // partial GAP over tile rows
        }
        psum += __shfl_xor(psum, 16, 32);              // fold the two M-half groups
        if (lane < 16) gap[wv * 64 + nt * 16 + n_l] = psum;
        __syncthreads();
    }
    if (t < 64) {
        float sum = 0.f;
        for (int wvi = 0; wvi < 8; ++wvi) sum += gap[wvi * 64 + t];
        feats[patch * 64 + t] = sum * (1.0f / 256.0f) + b2[t];  // GAP mean + bias
    }
}

// ---------------- kernel 4: mean over 16 patches -> d_out -------------------
__global__ void final_kernel(const float* __restrict__ feats, float* __restrict__ out) {
    int i = blockIdx.x * blockDim.x + threadIdx.x;
    if (i >= BATCH * 64) return;
    int b = i >> 6, oc = i & 63;
    float s = 0.f;
    for (int k = 0; k < KSEL; ++k) s += feats[(b * KSEL + k) * 64 + oc];
    out[i] = s * (1.0f / 16.0f);
}

extern "C" void kernel_launch(void* const* d_in, const int* in_sizes, int n_in,
                              void* d_out, int out_size, void* d_ws, size_t ws_size,
                              hipStream_t stream) {
    (void)in_sizes; (void)n_in; (void)out_size; (void)ws_size;
    const float* x  = (const float*)d_in[0];
    const float* w1 = (const float*)d_in[1];
    const float* b1 = (const float*)d_in[2];
    const float* w2 = (const float*)d_in[3];
    const float* b2 = (const float*)d_in[4];
    float* out = (float*)d_out;

    char*  ws     = (char*)d_ws;
    float* var    = (float*)ws;                                          // 70688 f32
    int*   coords = (int*)(ws + (size_t)BATCH * LWIN * sizeof(float));   // 1024 i32
    float* feats  = (float*)(ws + (size_t)BATCH * LWIN * sizeof(float)
                                + (size_t)NPATCH * 2 * sizeof(int));     // 32768 f32

    var_kernel<<<(BATCH * LWIN + 7) / 8, 256, 0, stream>>>(x, var);
    topk_kernel<<<BATCH, 256, 0, stream>>>(var, coords);
    size_t shbytes = (size_t)(INS_N + Y1_N + W2P_N + 864 + 32 + 512) * sizeof(float);
    encode_kernel<<<NPATCH, 256, shbytes, stream>>>(x, w1, b1, w2, b2, coords, feats);
    final_kernel<<<(BATCH * 64 + 255) / 256, 256, 0, stream>>>(feats, out);
}